// LocalityAdaptive_56427280335404
// MI455X (gfx1250) — compile-verified
//
#include <hip/hip_runtime.h>
#include <stdint.h>

// Problem constants (fixed by the reference setup).
#define ROI    32
#define DIM    192         // D == H == W
#define NBLK   6           // DIM / ROI
#define NCH    3           // channels C
#define NBATCH 4           // N
#define DPAIRS (DIM / 2)   // 96: grid.z covers N * DPAIRS, 2 d-slices per block

// out[n,d,h,w] = sum_c x[n,d,h,w,c] * K[d/32,h/32,w/32,c]   (F == 1)
//
// Grid:  (NBLK, NBLK, NBATCH*DPAIRS) -> one workgroup per 32(w) x 32(h) x 2(d)
//        slab; both d-slices share one ROI depth-block, so the 3 weights are
//        workgroup-uniform.
// Block: 256 threads; thread t -> 8 voxels along W (96 B of x, two float4 out).
// x is staged global->LDS with CDNA5 async-to-LDS b128 ops (ASYNCcnt), then
// consumed with ds_load_b128.
__global__ __launch_bounds__(256)
void la_block_conv_kernel(const float* __restrict__ x,
                          const float* __restrict__ kern,
                          float* __restrict__ out) {
    __shared__ char smem[256 * 96];   // 24 KB staging buffer

    const unsigned t  = threadIdx.x;
    const unsigned bz = blockIdx.z;              // n*DPAIRS + dpair
    const unsigned n  = bz / DPAIRS;
    const unsigned dp = bz - n * DPAIRS;         // d pair index
    const unsigned bh = blockIdx.y;
    const unsigned bw = blockIdx.x;

    const unsigned d  = dp * 2 + (t >> 7);       // waves 0-3: d0, waves 4-7: d0+1
    const unsigned bd = (dp * 2) >> 5;           // uniform across both slices

    // Workgroup-uniform weights (3 floats) -> scalar load + broadcast.
    const float* kp = kern + ((size_t)((bd * NBLK + bh) * NBLK + bw)) * NCH;
    const float k0 = kp[0];
    const float k1 = kp[1];
    const float k2 = kp[2];

    // Thread -> (h within block, 8-voxel group along w)
    const unsigned h = bh * ROI + ((t >> 2) & 31u);
    const unsigned w = bw * ROI + ((t & 3u) << 3);
    const size_t vox = (((size_t)(n * DIM + d) * DIM + h) * DIM) + w;

    // 8 voxels * 3 channels = 24 floats = 96 bytes, 16B aligned (w % 8 == 0).
    const char* gp  = (const char*)(x + vox * NCH);
    unsigned    lds = (unsigned)(uintptr_t)(&smem[t * 96]);

    // CDNA5 async copy global -> LDS: the instruction offset is added to BOTH
    // the LDS and global addresses, so one base pair + offset:16k stages 96
    // contiguous bytes.
    asm volatile(
        "global_load_async_to_lds_b128 %0, %1, off\n\t"
        "global_load_async_to_lds_b128 %0, %1, off offset:16\n\t"
        "global_load_async_to_lds_b128 %0, %1, off offset:32\n\t"
        "global_load_async_to_lds_b128 %0, %1, off offset:48\n\t"
        "global_load_async_to_lds_b128 %0, %1, off offset:64\n\t"
        "global_load_async_to_lds_b128 %0, %1, off offset:80"
        :: "v"(lds), "v"(gp) : "memory");

    // Wait for the async transfers (ASYNCcnt). Each thread reads back only its
    // own chunk, so no workgroup barrier is required.
    asm volatile("s_wait_asynccnt 0" ::: "memory");

    const float4* sp = (const float4*)(&smem[t * 96]);
    const float4 s0 = sp[0];   // v0.c0 v0.c1 v0.c2 v1.c0
    const float4 s1 = sp[1];   // v1.c1 v1.c2 v2.c0 v2.c1
    const float4 s2 = sp[2];   // v2.c2 v3.c0 v3.c1 v3.c2
    const float4 s3 = sp[3];   // v4.c0 v4.c1 v4.c2 v5.c0
    const float4 s4 = sp[4];   // v5.c1 v5.c2 v6.c0 v6.c1
    const float4 s5 = sp[5];   // v6.c2 v7.c0 v7.c1 v7.c2

    float4 oa, ob;
    oa.x = s0.x * k0 + s0.y * k1 + s0.z * k2;
    oa.y = s0.w * k0 + s1.x * k1 + s1.y * k2;
    oa.z = s1.z * k0 + s1.w * k1 + s2.x * k2;
    oa.w = s2.y * k0 + s2.z * k1 + s2.w * k2;
    ob.x = s3.x * k0 + s3.y * k1 + s3.z * k2;
    ob.y = s3.w * k0 + s4.x * k1 + s4.y * k2;
    ob.z = s4.z * k0 + s4.w * k1 + s5.x * k2;
    ob.w = s5.y * k0 + s5.z * k1 + s5.w * k2;

    *(float4*)(out + vox)     = oa;   // F == 1 -> out index == voxel index
    *(float4*)(out + vox + 4) = ob;
}

extern "C" void kernel_launch(void* const* d_in, const int* in_sizes, int n_in,
                              void* d_out, int out_size, void* d_ws, size_t ws_size,
                              hipStream_t stream) {
    (void)in_sizes; (void)n_in; (void)out_size; (void)d_ws; (void)ws_size;
    const float* x    = (const float*)d_in[0];   // [N, D, H, W, C] f32
    const float* kern = (const float*)d_in[1];   // [Bd, Bh, Bw, C, F=1] f32
    float*       out  = (float*)d_out;           // [N, D, H, W, 1] f32

    dim3 grid(NBLK, NBLK, NBATCH * DPAIRS);
    dim3 block(256, 1, 1);
    la_block_conv_kernel<<<grid, block, 0, stream>>>(x, kern, out);
}